// ScaledDotProductAttention_7146825581200
// MI455X (gfx1250) — compile-verified
//
#include <hip/hip_runtime.h>

// ---------------------------------------------------------------------------
// Scaled dot-product attention (fwd, returns output AND attn) for MI455X.
//
// Roofline: attn writeback (536 MB fp32) dominates -> ~27us floor at 23.3TB/s.
// Compute (34.4 GFLOP) is far below that, so we keep full fp32 precision and
// use the native CDNA5 fp32 matrix op V_WMMA_F32_16X16X4_F32 (wave32).
// LDS (320KB/WGP on CDNA5) holds the whole 16x2048 score strip per q-block.
// ---------------------------------------------------------------------------

typedef float v2f __attribute__((ext_vector_type(2)));
typedef float v4f __attribute__((ext_vector_type(4)));
typedef float v8f __attribute__((ext_vector_type(8)));

#define ATT_B 2
#define ATT_H 16
#define ATT_S 2048
#define ATT_D 64
#define LDS_STRIDE (ATT_S + 4)   // +4 floats: de-conflict 64-bank LDS for K-major reads
#define NEG_INF -1000000000.0f
#define INV_TEMP 0.125f          // 1/8.0

__global__ __launch_bounds__(128) void attn_fp32_wmma_kernel(
    const float* __restrict__ qg, const float* __restrict__ kg,
    const float* __restrict__ vg, const int* __restrict__ maskg,
    float* __restrict__ outg, float* __restrict__ attng)
{
    extern __shared__ float smem[];                 // scores[16][LDS_STRIDE] + scratch
    float* scores = smem;                           // 16 * 2052 floats
    float* red    = smem + 16 * LDS_STRIDE;         // 128 floats (16 rows x 8 partials)
    float* rmaxs  = red + 128;                      // 16 floats
    float* rinvs  = rmaxs + 16;                     // 16 floats

    const int b     = blockIdx.z;
    const int h     = blockIdx.y;
    const int qbase = blockIdx.x * 16;
    const int tid   = threadIdx.x;
    const int wave  = tid >> 5;
    const int lane  = tid & 31;
    const int n16   = lane & 15;        // M (A) / N (B,C,D) index within tile
    const int half  = lane >> 4;        // 0: K={0,1} / rows 0-7 ; 1: K={2,3} / rows 8-15
    const int koff  = half * 2;

    const size_t headOff = ((size_t)(b * ATT_H + h)) * ATT_S * ATT_D;
    const float* Q = qg + headOff;
    const float* K = kg + headOff;
    const float* V = vg + headOff;
    const int*   M = maskg + (size_t)b * ATT_S * ATT_S;
    float* attnHead = attng + ((size_t)(b * ATT_H + h)) * ATT_S * ATT_S;

    // ---- Preload Q A-fragments (16x4 fp32 A layout, ISA 7.12.2), pre-scaled ----
    v2f qa[16];
    {
        const float* qrow = Q + (size_t)(qbase + n16) * ATT_D + koff;
#pragma unroll
        for (int s = 0; s < 16; ++s) {
            qa[s].x = qrow[4 * s + 0] * INV_TEMP;
            qa[s].y = qrow[4 * s + 1] * INV_TEMP;
        }
    }

    // ---- GEMM1: scores = (Q/T) @ K^T ; waves stripe over 128 kv tiles ----
    for (int t = wave; t < ATT_S / 16; t += 4) {
        const int col = t * 16 + n16;                       // kv index this lane serves
        const float* krow = K + (size_t)col * ATT_D + koff; // B[k][n] = K[col][d]
        v8f acc = {0.f, 0.f, 0.f, 0.f, 0.f, 0.f, 0.f, 0.f};
#pragma unroll
        for (int s = 0; s < 16; ++s) {
            v2f kb;
            kb.x = krow[4 * s + 0];
            kb.y = krow[4 * s + 1];
            acc = __builtin_amdgcn_wmma_f32_16x16x4_f32(
                false, qa[s], false, kb, (short)0, acc, false, false);
        }
        // mask + park in LDS. C layout: lanes0-15 rows 0-7, lanes16-31 rows 8-15.
#pragma unroll
        for (int r = 0; r < 8; ++r) {
            const int m = half * 8 + r;
            const int mv = M[(size_t)(qbase + m) * ATT_S + col];
            scores[m * LDS_STRIDE + col] = mv ? acc[r] : NEG_INF;
        }
    }
    __syncthreads();

    // ---- Softmax over each of the 16 rows (8 threads per row) ----
    const int row = tid >> 3;
    const int sub = tid & 7;
    {
        float mx = -INFINITY;
        for (int i = sub; i < ATT_S; i += 8)
            mx = fmaxf(mx, scores[row * LDS_STRIDE + i]);
        red[row * 8 + sub] = mx;
        __syncthreads();
        if (sub == 0) {
            float m2 = red[row * 8];
#pragma unroll
            for (int j = 1; j < 8; ++j) m2 = fmaxf(m2, red[row * 8 + j]);
            rmaxs[row] = m2;
        }
        __syncthreads();
        const float rm = rmaxs[row];
        float sum = 0.f;
        for (int i = sub; i < ATT_S; i += 8) {
            const float e = __expf(scores[row * LDS_STRIDE + i] - rm);
            scores[row * LDS_STRIDE + i] = e;
            sum += e;
        }
        red[row * 8 + sub] = sum;
        __syncthreads();
        if (sub == 0) {
            float s2 = 0.f;
#pragma unroll
            for (int j = 0; j < 8; ++j) s2 += red[row * 8 + j];
            rinvs[row] = 1.0f / s2;
        }
        __syncthreads();
    }

    // ---- Fused normalize + streaming attn writeback (NT b128 stores) ----
    for (int r = 0; r < 16; ++r) {
        const float inv = rinvs[r];
        v4f* src = (v4f*)(scores + r * LDS_STRIDE);
        v4f* dst = (v4f*)(attnHead + (size_t)(qbase + r) * ATT_S);
        for (int c = tid; c < ATT_S / 4; c += 128) {
            v4f p = src[c];
            p *= inv;
            src[c] = p;                              // normalized P stays in LDS for GEMM2
            __builtin_nontemporal_store(p, &dst[c]); // 536MB stream: don't pollute caches
        }
    }
    __syncthreads();

    // ---- GEMM2: out[16,64] = P[16,2048] @ V[2048,64]; each wave owns 16 cols of D ----
    {
        const int nbase = wave * 16;
        const float* vcol = V + nbase + n16;         // B[k][n] = V[kv][nbase+n16]
        const float* prow = scores + n16 * LDS_STRIDE + koff;
        v8f acc = {0.f, 0.f, 0.f, 0.f, 0.f, 0.f, 0.f, 0.f};
#pragma unroll 4
        for (int s = 0; s < ATT_S / 4; ++s) {
            const int kb = 4 * s;
            v2f pa;                                  // A-frag from LDS (b64, conflict-free)
            pa.x = prow[kb + 0];
            pa.y = prow[kb + 1];
            v2f vb;                                  // B-frag from global (L2-resident V)
            vb.x = vcol[(size_t)(kb + koff + 0) * ATT_D];
            vb.y = vcol[(size_t)(kb + koff + 1) * ATT_D];
            acc = __builtin_amdgcn_wmma_f32_16x16x4_f32(
                false, pa, false, vb, (short)0, acc, false, false);
        }
#pragma unroll
        for (int r = 0; r < 8; ++r) {
            const int m = half * 8 + r;
            outg[headOff + (size_t)(qbase + m) * ATT_D + nbase + n16] = acc[r];
        }
    }
}

extern "C" void kernel_launch(void* const* d_in, const int* in_sizes, int n_in,
                              void* d_out, int out_size, void* d_ws, size_t ws_size,
                              hipStream_t stream) {
    const float* q    = (const float*)d_in[0];
    const float* k    = (const float*)d_in[1];
    const float* v    = (const float*)d_in[2];
    const int*   mask = (const int*)d_in[3];

    float* outp  = (float*)d_out;                                   // [B,H,S,D]
    float* attnp = outp + (size_t)ATT_B * ATT_H * ATT_S * ATT_D;    // [B,H,S,S]

    const size_t shmem = (size_t)(16 * LDS_STRIDE + 128 + 32) * sizeof(float); // ~132 KB
    (void)hipFuncSetAttribute((const void*)attn_fp32_wmma_kernel,
                              hipFuncAttributeMaxDynamicSharedMemorySize, (int)shmem);

    dim3 grid(ATT_S / 16, ATT_H, ATT_B);   // 128 x 16 x 2 = 4096 workgroups
    attn_fp32_wmma_kernel<<<grid, 128, shmem, stream>>>(q, k, v, mask, outp, attnp);
}